// ProposalLayer_23553600652004
// MI455X (gfx1250) — compile-verified
//
#include <hip/hip_runtime.h>
#include <hip/hip_bf16.h>
#include <math.h>

// ---------------- problem constants (from reference) ----------------
#define FEAT_A   18675        // 83 * 25 * 9 anchors
#define N_PRE    12000
#define N_POST   2000
#define IMGX     1333.0f
#define IMGY     402.0f
#define MINSZ    16.0f
#define NMS_T    0.7f

// ---------------- workspace layout (float offsets) ----------------
#define SC_OFF   0            // FEAT_A floats  masked scores
#define ORD_OFF  18688        // N_PRE ints     top-N_PRE anchor indices (score-desc, stable)
#define BX_OFF   30688        // N_PRE float4   decoded boxes (16B aligned: 30688*4 % 16 == 0)
#define MK_OFF   78688        // N_PRE ints     suppression marks
// total = 90688 floats = 362,752 bytes of d_ws

// ---------------- CDNA5 async global->LDS helpers (ASYNCcnt path) ----------
__device__ __forceinline__ void async_copy_b128(const float4* gsrc, float4* ldst) {
    unsigned ldsaddr = (unsigned)(size_t)ldst;   // low 32b of flat addr == LDS byte offset
    asm volatile("global_load_async_to_lds_b128 %0, %1, off"
                 :: "v"(ldsaddr), "v"(gsrc) : "memory");
}
__device__ __forceinline__ void async_copy_b32(const float* gsrc, float* ldst) {
    unsigned ldsaddr = (unsigned)(size_t)ldst;
    asm volatile("global_load_async_to_lds_b32 %0, %1, off"
                 :: "v"(ldsaddr), "v"(gsrc) : "memory");
}
__device__ __forceinline__ void async_wait0() {
    asm volatile("s_wait_asynccnt 0" ::: "memory");
}

// ---------------- shared decode math (matches _decode) ----------------
__device__ __forceinline__ float4 decode_box(float4 a, float4 d) {
    float w_a  = a.z - a.x;
    float h_a  = a.w - a.y;
    float cx_a = a.x + 0.5f * w_a;
    float cy_a = a.y + 0.5f * h_a;
    float cx = d.x * w_a + cx_a;
    float cy = d.y * h_a + cy_a;
    float w  = expf(d.z) * w_a;
    float h  = expf(d.w) * h_a;
    float4 b;
    b.x = fminf(fmaxf(cx - 0.5f * w, 0.0f), IMGX);   // x1
    b.z = fminf(fmaxf(cx + 0.5f * w, 0.0f), IMGX);   // x2
    b.y = fminf(fmaxf(cy - 0.5f * h, 0.0f), IMGY);   // y1
    b.w = fminf(fmaxf(cy + 0.5f * h, 0.0f), IMGY);   // y2
    return b;
}

// ---------------- kernel 0: zero the whole output ----------------
__global__ void k_zero(float* __restrict__ out, int n) {
    int i = blockIdx.x * 256 + threadIdx.x;
    if (i < n) out[i] = 0.0f;
}

// ---------------- kernel 1: decode validity + masked score ----------------
__global__ void k_score(const float4* __restrict__ anch,
                        const float*  __restrict__ cls,
                        const float4* __restrict__ pred,
                        float* __restrict__ sc) {
    int i = blockIdx.x * 256 + threadIdx.x;
    if (i >= FEAT_A) return;
    float4 b = decode_box(anch[i], pred[i]);
    bool valid = (b.w - b.y >= MINSZ) && (b.z - b.x >= MINSZ);
    float s = cls[2 * i + 1];
    sc[i] = valid ? s : -__builtin_inff();
}

// ---------------- kernel 2: stable descending rank (counting sort) ----------
// rank[i] = #{j : s[j] > s[i]} + #{j < i : s[j] == s[i]}  (== jnp stable argsort).
// Ranks are a permutation of 0..A-1, so each ord[0..N_PRE) slot is written once.
// Double-buffered async LDS staging; threads whose rank already exceeds N_PRE
// skip further counting (their result can never be stored).
__global__ void k_rank(const float* __restrict__ sc, int* __restrict__ ord) {
    __shared__ float tile[2][256];
    const int t = threadIdx.x;
    const int i = blockIdx.x * 256 + t;
    float si = (i < FEAT_A) ? sc[i] : 0.0f;
    int cnt = 0;

    const int ntiles = (FEAT_A + 255) / 256;     // 73
    // prologue: stage tile 0 into buffer 0
    if (t < 256 && t < FEAT_A) async_copy_b32(sc + t, &tile[0][t]);
    async_wait0();
    __syncthreads();

    for (int tt = 0; tt < ntiles; ++tt) {
        const int cur = tt & 1;
        // prefetch next tile into the other buffer (its readers finished at the
        // barrier that ended the previous iteration)
        if (tt + 1 < ntiles) {
            int nb = (tt + 1) * 256;
            int nn = min(256, FEAT_A - nb);
            if (t < nn) async_copy_b32(sc + nb + t, &tile[1 - cur][t]);
        }
        // compute on current tile
        int base = tt * 256;
        int n = min(256, FEAT_A - base);
        if (cnt < N_PRE) {
            const float4* t4 = (const float4*)tile[cur];
            int nq = n >> 2;
            for (int q = 0; q < nq; ++q) {
                float4 s4 = t4[q];
                int j = base + 4 * q;
                cnt += ((s4.x > si) || (s4.x == si && (j + 0) < i)) ? 1 : 0;
                cnt += ((s4.y > si) || (s4.y == si && (j + 1) < i)) ? 1 : 0;
                cnt += ((s4.z > si) || (s4.z == si && (j + 2) < i)) ? 1 : 0;
                cnt += ((s4.w > si) || (s4.w == si && (j + 3) < i)) ? 1 : 0;
            }
            for (int jj = nq << 2; jj < n; ++jj) {
                float sj = tile[cur][jj];
                int j = base + jj;
                cnt += ((sj > si) || (sj == si && j < i)) ? 1 : 0;
            }
        }
        async_wait0();       // next tile's async copy has landed in LDS
        __syncthreads();
    }
    if (i < FEAT_A && cnt < N_PRE) ord[cnt] = i;
}

// ---------------- kernel 3: gather + re-decode the selected boxes ----------
__global__ void k_gather(const float4* __restrict__ anch,
                         const float4* __restrict__ pred,
                         const int* __restrict__ ord,
                         float4* __restrict__ bx) {
    int k = blockIdx.x * 256 + threadIdx.x;
    if (k >= N_PRE) return;
    int i = ord[k];
    bx[k] = decode_box(anch[i], pred[i]);
}

// ---------------- kernel 4: parallel matrix-NMS marks ----------------
// mark[j] = OR over z<j of (IoU(z,j) >= NMS_T).  The reference's scan never
// consults mark inside row(z), so this is fully parallel.  z-tiles staged via
// double-buffered CDNA5 async global->LDS copies.
__global__ void k_nms(const float4* __restrict__ bx, int* __restrict__ mark) {
    __shared__ float4 tile[2][256];
    const int t = threadIdx.x;
    const int j = blockIdx.x * 256 + t;
    float4 bj = bx[min(j, N_PRE - 1)];
    float aj = (bj.z - bj.x + 1.0f) * (bj.w - bj.y + 1.0f);
    bool mk = false;

    const int ntiles = blockIdx.x + 1;           // tiles 0..blockIdx.x
    // prologue: stage tile 0 into buffer 0 (always full: zbase 0, N_PRE > 256)
    async_copy_b128(bx + t, &tile[0][t]);
    async_wait0();
    __syncthreads();

    for (int tt = 0; tt < ntiles; ++tt) {
        const int cur = tt & 1;
        if (tt + 1 < ntiles) {
            int nb = (tt + 1) * 256;
            int nn = min(256, N_PRE - nb);
            if (t < nn) async_copy_b128(bx + nb + t, &tile[1 - cur][t]);
        }
        int zbase = tt * 256;
        int nz = min(256, N_PRE - zbase);
        if (!mk && j < N_PRE) {
            int lim = min(nz, j - zbase);        // only z < j suppress j
            for (int zz = 0; zz < lim; ++zz) {
                float4 bz = tile[cur][zz];
                float xx1 = fmaxf(bz.x, bj.x);
                float yy1 = fmaxf(bz.y, bj.y);
                float xx2 = fminf(bz.z, bj.z);
                float yy2 = fminf(bz.w, bj.w);
                float w = fmaxf(0.0f, xx2 - xx1 + 1.0f);
                float h = fmaxf(0.0f, yy2 - yy1 + 1.0f);
                float inter = w * h;
                float az = (bz.z - bz.x + 1.0f) * (bz.w - bz.y + 1.0f);
                // inter/(az+aj-inter) >= T  <=>  inter*(1+T) >= T*(az+aj)
                mk = mk || (inter * (1.0f + NMS_T) >= NMS_T * (az + aj));
            }
        }
        async_wait0();
        __syncthreads();
    }
    if (j < N_PRE) mark[j] = mk ? 1 : 0;
}

// ---------------- kernel 5: stable compaction of kept boxes ----------------
// Matches argsort((~keep).astype(i32), stable)[:N_POST] + where(kept, boxes, 0):
// first K kept boxes in score order get (box, flag=1); everything else stays 0.
__global__ void k_emit(const float4* __restrict__ bx,
                       const int* __restrict__ mark,
                       float* __restrict__ out) {
    __shared__ int cnts[256];
    __shared__ int offs[256];
    const int t = threadIdx.x;
    const int SEG = (N_PRE + 255) / 256;          // 47
    int j0 = t * SEG;
    int j1 = min(j0 + SEG, N_PRE);
    int c = 0;
    for (int j = j0; j < j1; ++j) c += (mark[j] == 0) ? 1 : 0;
    cnts[t] = c;
    __syncthreads();
    if (t == 0) {
        int run = 0;
        for (int k = 0; k < 256; ++k) { offs[k] = run; run += cnts[k]; }
    }
    __syncthreads();
    int pos = offs[t];
    for (int j = j0; j < j1; ++j) {
        if (mark[j] == 0) {
            if (pos < N_POST) {
                float4 b = bx[j];
                out[4 * pos + 0] = b.x;
                out[4 * pos + 1] = b.y;
                out[4 * pos + 2] = b.z;
                out[4 * pos + 3] = b.w;
                out[4 * N_POST + pos] = 1.0f;     // kept flag
            }
            ++pos;
        }
    }
}

// ---------------- launch ----------------
extern "C" void kernel_launch(void* const* d_in, const int* in_sizes, int n_in,
                              void* d_out, int out_size, void* d_ws, size_t ws_size,
                              hipStream_t stream) {
    const float4* anch = (const float4*)d_in[0];   // (A,4)   f32
    const float*  cls  = (const float*) d_in[1];   // (1,A,2) f32
    const float4* pred = (const float4*)d_in[2];   // (1,A,4) f32

    float*  ws_f = (float*)d_ws;
    float*  sc   = ws_f + SC_OFF;
    int*    ord  = (int*)   (ws_f + ORD_OFF);
    float4* bx   = (float4*)(ws_f + BX_OFF);
    int*    mark = (int*)   (ws_f + MK_OFF);
    float*  out  = (float*)d_out;                  // [2000*4 rois | 2000 kept]

    (void)in_sizes; (void)n_in; (void)ws_size;

    k_zero  <<<(out_size + 255) / 256, 256, 0, stream>>>(out, out_size);
    k_score <<<(FEAT_A  + 255) / 256, 256, 0, stream>>>(anch, cls, pred, sc);
    k_rank  <<<(FEAT_A  + 255) / 256, 256, 0, stream>>>(sc, ord);
    k_gather<<<(N_PRE   + 255) / 256, 256, 0, stream>>>(anch, pred, ord, bx);
    k_nms   <<<(N_PRE   + 255) / 256, 256, 0, stream>>>(bx, mark);
    k_emit  <<<1, 256, 0, stream>>>(bx, mark, out);
}